// Entropy_43508018708562
// MI455X (gfx1250) — compile-verified
//
#include <hip/hip_runtime.h>
#include <hip/hip_bf16.h>
#include <math.h>

// Problem constants (reference: x of shape (32, 3, 512, 512) f32)
#define NBINS          256
#define HW             (512 * 512)        // 262144 pixels per channel plane
#define IMG_STRIDE     (3 * HW)           // floats per image
#define BLOCKS_PER_IMG 32
#define THREADS        256
// Each block: HW / BLOCKS_PER_IMG = 8192 pixels = 2048 float4 groups per channel
#define V4_PER_BLOCK   (HW / BLOCKS_PER_IMG / 4)   // 2048
#define ITERS          (V4_PER_BLOCK / THREADS)    // 8

// ---------------------------------------------------------------------------
// Kernel 0: zero the per-image global histograms in workspace (deterministic
// re-init every launch; we accumulate with atomics below).
// ---------------------------------------------------------------------------
__global__ void zero_hist_kernel(unsigned int* __restrict__ ghist) {
    ghist[blockIdx.x * NBINS + threadIdx.x] = 0u;
}

// ---------------------------------------------------------------------------
// CDNA5 async-copy helpers.
// LDS aperture: generic (flat) address of a __shared__ object carries the
// LDS byte offset in its low 32 bits, which is exactly what the VDST operand
// of GLOBAL_LOAD_ASYNC_TO_LDS expects (a byte offset added to the wave's LDS base).
// ---------------------------------------------------------------------------
__device__ __forceinline__ unsigned lds_addr_of(const void* p) {
    return (unsigned)(unsigned long long)(uintptr_t)p;
}

// One 16-byte DMA: global memory -> LDS, tracked by ASYNCcnt (not LOADcnt).
__device__ __forceinline__ void async_copy_b128(unsigned lds_addr, const void* gptr) {
    asm volatile("global_load_async_to_lds_b128 %0, %1, off"
                 :
                 : "v"(lds_addr), "v"((unsigned long long)(uintptr_t)gptr)
                 : "memory");
}

// ---------------------------------------------------------------------------
// Kernel 1: streaming luminance histogram with async-to-LDS double buffering.
// grid = 32 images * 32 blocks, block = 256 threads (8 wave32 waves).
//
// Data path per iteration (3 channels x 16B per thread):
//   HBM --global_load_async_to_lds_b128--> LDS stage buffer  (ASYNCcnt)
//   LDS --ds_load_b128--> VGPR --VALU--> bin --ds_add_u32--> LDS histogram
// Tile i+1 streams over HBM while tile i is histogrammed.
//
// Per-LANE privatized LDS histogram: lhist[bin*32 + lane].
//   Stride 32 => within a wave, lanes can never bank-conflict on updates:
//   (32*bin1 + l1) == (32*bin2 + l2) mod 64 forces l1 == l2.
//   atomicAdd with unused return lowers to ds_add_u32 (no-return LDS atomic),
//   contended only across the 8 waves of the block.
// ---------------------------------------------------------------------------
__global__ __launch_bounds__(THREADS)
void hist_kernel(const float* __restrict__ x, unsigned int* __restrict__ ghist) {
    __shared__ unsigned int lhist[NBINS * 32];     // 32 KB
    __shared__ float4 stage[2][3][THREADS];        // 24 KB double-buffered tiles

    const int tid  = threadIdx.x;
    const int lane = tid & 31;

    // Zero LDS histogram
    #pragma unroll
    for (int i = 0; i < (NBINS * 32) / THREADS; ++i)
        lhist[i * THREADS + tid] = 0u;
    __syncthreads();

    const int img = blockIdx.x >> 5;       // / BLOCKS_PER_IMG
    const int blk = blockIdx.x & 31;       // % BLOCKS_PER_IMG

    const size_t base = (size_t)img * IMG_STRIDE;
    const float4* __restrict__ Rv = (const float4*)(x + base);
    const float4* __restrict__ Gv = (const float4*)(x + base + HW);
    const float4* __restrict__ Bv = (const float4*)(x + base + 2 * HW);

    const int start = blk * V4_PER_BLOCK + tid;

    // Prologue: issue DMA for tile 0 into buffer 0 (ASYNCcnt = 3).
    async_copy_b128(lds_addr_of(&stage[0][0][tid]), &Rv[start]);
    async_copy_b128(lds_addr_of(&stage[0][1][tid]), &Gv[start]);
    async_copy_b128(lds_addr_of(&stage[0][2][tid]), &Bv[start]);

    #pragma unroll
    for (int it = 0; it < ITERS; ++it) {
        const int b = it & 1;

        if (it + 1 < ITERS) {
            // Issue DMA for the next tile into the other buffer, then wait
            // until only those 3 ops remain in flight. Async loads complete
            // in order per wave, so ASYNCcnt <= 3 proves tile `it` landed.
            const int v  = start + (it + 1) * THREADS;
            const int nb = b ^ 1;
            async_copy_b128(lds_addr_of(&stage[nb][0][tid]), &Rv[v]);
            async_copy_b128(lds_addr_of(&stage[nb][1][tid]), &Gv[v]);
            async_copy_b128(lds_addr_of(&stage[nb][2][tid]), &Bv[v]);
            asm volatile("s_wait_asynccnt 0x3" ::: "memory");
        } else {
            asm volatile("s_wait_asynccnt 0x0" ::: "memory");
        }

        // Read back this thread's own staged slots (no cross-thread sharing,
        // so no barrier needed; per-wave ASYNCcnt wait is sufficient).
        const float4 r  = stage[b][0][tid];   // ds_load_b128
        const float4 g  = stage[b][1][tid];
        const float4 bl = stage[b][2][tid];

        float yv[4];
        yv[0] = fmaf(0.114f, bl.x, fmaf(0.587f, g.x, 0.299f * r.x)) * 255.0f;
        yv[1] = fmaf(0.114f, bl.y, fmaf(0.587f, g.y, 0.299f * r.y)) * 255.0f;
        yv[2] = fmaf(0.114f, bl.z, fmaf(0.587f, g.z, 0.299f * r.z)) * 255.0f;
        yv[3] = fmaf(0.114f, bl.w, fmaf(0.587f, g.w, 0.299f * r.w)) * 255.0f;

        #pragma unroll
        for (int k = 0; k < 4; ++k) {
            // jnp.round == round-half-to-even == rintf (default RNE mode)
            float yr = rintf(yv[k]);
            yr = fminf(fmaxf(yr, 0.0f), 255.0f);
            const int bin = (int)yr;
            atomicAdd(&lhist[bin * 32 + lane], 1u);   // ds_add_u32
        }
    }
    __syncthreads();

    // Reduce the 32 lane-private sub-histograms; one global atomic per bin.
    // tid == bin index (THREADS == NBINS).
    unsigned int sum = 0u;
    #pragma unroll
    for (int l = 0; l < 32; ++l)
        sum += lhist[tid * 32 + l];
    atomicAdd(&ghist[img * NBINS + tid], sum);        // global_atomic_add_u32
}

// ---------------------------------------------------------------------------
// Kernel 2: entropy per image. 32 blocks x 256 threads.
// p = count / 262144 (total is exact: every pixel lands in [0,255]).
// ---------------------------------------------------------------------------
__global__ __launch_bounds__(THREADS)
void entropy_kernel(const unsigned int* __restrict__ ghist,
                    float* __restrict__ out) {
    __shared__ float red[THREADS];
    const int img = blockIdx.x;
    const int t   = threadIdx.x;

    const float p    = (float)ghist[img * NBINS + t] * (1.0f / (float)HW);
    const float term = -p * log2f(p + 1e-8f);

    red[t] = term;
    __syncthreads();
    #pragma unroll
    for (int s = THREADS / 2; s > 0; s >>= 1) {
        if (t < s) red[t] += red[t + s];
        __syncthreads();
    }
    if (t == 0) out[img] = red[0];
}

// ---------------------------------------------------------------------------
extern "C" void kernel_launch(void* const* d_in, const int* in_sizes, int n_in,
                              void* d_out, int out_size, void* d_ws, size_t ws_size,
                              hipStream_t stream) {
    const float* x = (const float*)d_in[0];
    float* out = (float*)d_out;
    unsigned int* ghist = (unsigned int*)d_ws;       // 32*256*4 = 32 KB scratch

    const int n_img = in_sizes[0] / IMG_STRIDE;      // 32

    zero_hist_kernel<<<n_img, NBINS, 0, stream>>>(ghist);
    hist_kernel<<<n_img * BLOCKS_PER_IMG, THREADS, 0, stream>>>(x, ghist);
    entropy_kernel<<<n_img, THREADS, 0, stream>>>(ghist, out);
}